// Model_6725918786261
// MI455X (gfx1250) — compile-verified
//
#include <hip/hip_runtime.h>

// ---------------------------------------------------------------------------
// out = softmax(q5 + k4^T * (1/sqrt(56)) + mask) @ v2
// q5,k4 [B,H,S,S]; v2 [B,H,S,D]; mask [1,S,S]; out [B,H,S,D]; B=256 H=16 S=56 D=64.
// Memory-bound (~220MB @ 23.3TB/s ~ 9.4us). One workgroup per (b,h).
// P@V via V_WMMA_F32_16X16X4_F32 (exact fp32, two independent acc chains/wave).
// V tile staged with async global->LDS (ASYNCcnt) overlapped with softmax.
// ---------------------------------------------------------------------------

typedef __attribute__((ext_vector_type(2))) float v2f;
typedef __attribute__((ext_vector_type(8))) float v8f;
typedef int v4i __attribute__((vector_size(16)));   // matches async builtin param

#define SS   56
#define DV   64
#define NBH  4096                      // B*H
#define SCALE 0.13363062095621219f    // 1/sqrt(56)

#define KS 57                          // sK row stride (odd -> conflict-free transposed read)
#define PS 66                          // sP row stride (even -> 8B-aligned float2 A-frag loads)

#if __has_builtin(__builtin_amdgcn_global_load_async_to_lds_b128)
#define USE_ASYNC_V 1
#define GLOBAL_AS __attribute__((address_space(1)))
#define LDS_AS    __attribute__((address_space(3)))
#else
#define USE_ASYNC_V 0
#endif

__global__ __launch_bounds__(256)
void attn_pv_wmma(const float* __restrict__ q5,
                  const float* __restrict__ k4,
                  const float* __restrict__ v2,
                  const float* __restrict__ mask,
                  float* __restrict__ out)
{
    __shared__ float sK[SS * KS];      // k tile, row-major (read transposed)
    __shared__ float sP[64 * PS];      // softmax probs, zero-padded to 64x64
    __shared__ float sV[64 * DV];      // v tile, rows 56..63 zeroed

    const int tid  = threadIdx.x;
    const int wid  = tid >> 5;         // wave id (wave32 on gfx1250)
    const int lane = tid & 31;
    const int bh   = blockIdx.x;       // b*H + h

    const float* qg = q5  + (size_t)bh * (SS * SS);
    const float* kg = k4  + (size_t)bh * (SS * SS);
    const float* vg = v2  + (size_t)bh * (SS * DV);
    float*       og = out + (size_t)bh * (SS * DV);

    // ---- Phase 0: fire async V-tile loads (consumed only in phase 3) ------
    const float4* vg4 = (const float4*)vg;               // 896 vec4, rows of 16
#if USE_ASYNC_V
    for (int f = tid; f < (SS * DV) / 4; f += 256) {
        const int row = f >> 4;
        const int c4  = f & 15;
        __builtin_amdgcn_global_load_async_to_lds_b128(
            (GLOBAL_AS v4i*)(vg4 + f),
            (LDS_AS v4i*)&sV[row * DV + c4 * 4],
            /*offset=*/0, /*cpol=*/0);
    }
#endif

    // ---- Phase 1: stage K tile into LDS (transposed read later), zero pads
    {
        const float4* kg4 = (const float4*)kg;           // 784 vec4, rows of 14
        for (int f = tid; f < (SS * SS) / 4; f += 256) {
            float4 val = kg4[f];
            int row = f / (SS / 4);
            int col = (f % (SS / 4)) * 4;
            float* d = &sK[row * KS + col];
            d[0] = val.x; d[1] = val.y; d[2] = val.z; d[3] = val.w;
        }
#if !USE_ASYNC_V
        for (int f = tid; f < (SS * DV) / 4; f += 256) {
            float4 val = vg4[f];
            *(float4*)&sV[(f >> 4) * DV + (f & 15) * 4] = val;
        }
#endif
        for (int i = tid; i < (64 - SS) * DV; i += 256)  // V pad rows = 0
            sV[SS * DV + i] = 0.0f;
        for (int i = tid; i < (64 - SS) * PS; i += 256)  // P pad rows = 0
            sP[SS * PS + i] = 0.0f;
    }
    __syncthreads();

    // ---- Phase 2: row softmax of (q + k^T*scale + mask) -------------------
    // One wave per row; lane j covers columns j and j+32. Overlaps async V.
    for (int i = wid; i < SS; i += 8) {
        const int j1 = lane, j2 = lane + 32;
        float s1 = qg[i * SS + j1] + sK[j1 * KS + i] * SCALE + mask[i * SS + j1];
        float s2 = -INFINITY;
        if (j2 < SS)
            s2 = qg[i * SS + j2] + sK[j2 * KS + i] * SCALE + mask[i * SS + j2];

        float m = fmaxf(s1, s2);
        #pragma unroll
        for (int off = 16; off > 0; off >>= 1)
            m = fmaxf(m, __shfl_xor(m, off, 32));

        float e1 = __expf(s1 - m);
        float e2 = (j2 < SS) ? __expf(s2 - m) : 0.0f;
        float sum = e1 + e2;
        #pragma unroll
        for (int off = 16; off > 0; off >>= 1)
            sum += __shfl_xor(sum, off, 32);

        const float inv = 1.0f / sum;
        sP[i * PS + j1] = e1 * inv;
        sP[i * PS + j2] = e2 * inv;   // cols 56..63 get 0 -> K-dim padding
    }

#if USE_ASYNC_V
    // V tile must have landed in LDS before any wave reads it in phase 3.
#if __has_builtin(__builtin_amdgcn_s_wait_asynccnt)
    __builtin_amdgcn_s_wait_asynccnt(0);
#else
    asm volatile("s_wait_asynccnt 0x0" ::: "memory");
#endif
#endif
    __syncthreads();

    // ---- Phase 3: O = P(64x64) x V(64x64) via WMMA f32 16x16x4 ------------
    // Wave wid owns tiles (m0, n) and (m0+2, n): same B fragment, two
    // independent accumulator chains. Wave-uniform control flow (EXEC all-1s).
    const int lrow  = lane & 15;
    const int khalf = (lane < 16) ? 0 : 2;   // fp32 A/B lane-half K offset
    const int m0  = wid >> 2;                // 0 or 1 -> rows 0..31 (never OOB)
    const int nT  = wid & 3;
    const int ra0 = m0 * 16 + lrow;          // A row, tile 0
    const int ra1 = ra0 + 32;                // A row, tile 1 (m0+2)
    const int cb  = nT * 16 + lrow;          // B column / D column

    v8f c0 = {}, c1 = {};
    #pragma unroll
    for (int kk = 0; kk < 16; ++kk) {
        const int kc = kk * 4 + khalf;
        // A (16x4 f32): VGPR0 = K{0|2}, VGPR1 = K{1|3} -> float2 LDS load
        v2f a0 = *(const v2f*)&sP[ra0 * PS + kc];
        v2f a1 = *(const v2f*)&sP[ra1 * PS + kc];
        // B (4x16 f32): column per lane, rows kc and kc+1 (shared by tiles)
        v2f b;
        b.x = sV[kc * DV + cb];
        b.y = sV[(kc + 1) * DV + cb];
        c0 = __builtin_amdgcn_wmma_f32_16x16x4_f32(
                false, a0, false, b, (short)0, c0, false, false);
        c1 = __builtin_amdgcn_wmma_f32_16x16x4_f32(
                false, a1, false, b, (short)0, c1, false, false);
    }

    // D layout: VGPR r -> row base + r (lanes 0..15) / base + 8 + r (16..31)
    const int rb0 = m0 * 16 + ((lane < 16) ? 0 : 8);   // rows 0..31: no guard
    const int rb1 = rb0 + 32;                           // rows 32..63: guard
    #pragma unroll
    for (int r = 0; r < 8; ++r)
        og[(rb0 + r) * DV + cb] = c0[r];
    #pragma unroll
    for (int r = 0; r < 8; ++r) {
        const int row = rb1 + r;
        if (row < SS)
            og[row * DV + cb] = c1[r];
    }
}

extern "C" void kernel_launch(void* const* d_in, const int* in_sizes, int n_in,
                              void* d_out, int out_size, void* d_ws, size_t ws_size,
                              hipStream_t stream)
{
    const float* q5   = (const float*)d_in[0];
    const float* k4   = (const float*)d_in[1];
    const float* v2   = (const float*)d_in[2];
    const float* mask = (const float*)d_in[3];
    float* out = (float*)d_out;
    (void)in_sizes; (void)n_in; (void)out_size; (void)d_ws; (void)ws_size;

    attn_pv_wmma<<<NBH, 256, 0, stream>>>(q5, k4, v2, mask, out);
}